// StaticSusceptibilityGNN_15985868275842
// MI455X (gfx1250) — compile-verified
//
#include <hip/hip_runtime.h>
#include <hip/hip_bf16.h>

typedef float v2f __attribute__((ext_vector_type(2)));
typedef float v8f __attribute__((ext_vector_type(8)));

#define HID 64

// ---------------------------------------------------------------- zero
__global__ void zero_kernel(float4* __restrict__ p, long long n4) {
  long long i = blockIdx.x * (long long)blockDim.x + threadIdx.x;
  if (i < n4) p[i] = make_float4(0.f, 0.f, 0.f, 0.f);
}

// ------------------------------------------------- edge scatter (mean num/den)
// thread = (edge, channel); atomics land in L2 (feature tables are L2-resident)
template <int C, bool DO_CNT>
__global__ void scatter_kernel(const float* __restrict__ feat,
                               const int* __restrict__ src,
                               const int* __restrict__ dst,
                               float* __restrict__ agg,
                               float* __restrict__ cnt,
                               long long E) {
  long long t = blockIdx.x * (long long)blockDim.x + threadIdx.x;
  long long total = E * (long long)C;
  if (t >= total) return;
  long long e = t / C;               // C is a power of two -> shift
  int c = (int)(t - e * C);
  int s = src[e];
  int d = dst[e];
  atomicAdd(&agg[(long long)d * C + c], feat[(long long)s * C + c]);
  if (DO_CNT && c == 0) atomicAdd(&cnt[d], 1.0f);
}

// ------------------------------------------- SAGE transform via f32 WMMA
// out[n,:] = rownorm( mean[n,:] @ Wl + x[n,:] @ Wr + b ), optional ReLU
// One wave = 16 node rows x 64 cols (4 C-tiles), K stepped by 4 (16x16x4 f32).
template <int CIN, bool RELU>
__global__ void sage_wmma_kernel(const float* __restrict__ agg,
                                 const float* __restrict__ cnt,
                                 const float* __restrict__ xin,
                                 const float* __restrict__ Wl,
                                 const float* __restrict__ Wr,
                                 const float* __restrict__ bias,
                                 float* __restrict__ out, int N) {
  const int lane = threadIdx.x & 31;
  const int wid  = (blockIdx.x * blockDim.x + threadIdx.x) >> 5;
  const int m0   = wid * 16;
  if (m0 >= N) return;               // wave-uniform: EXEC stays all-1 for WMMA

  const int half = lane >> 4;        // 0: K/rows 0..1 half, 1: K 2..3 half
  const int l16  = lane & 15;

  int rowA = m0 + l16;               // A-matrix row owned by this lane
  if (rowA > N - 1) rowA = N - 1;    // clamp loads only; stores guarded later

  const float inv = 1.0f / fmaxf(cnt[rowA], 1.0f);
  const float* aggRow = agg + (long long)rowA * CIN;
  const float* xRow   = xin + (long long)rowA * CIN;

  v8f acc[4];
#pragma unroll
  for (int ct = 0; ct < 4; ++ct) {   // fold bias into accumulator init
    const float bv = bias[ct * 16 + l16];
#pragma unroll
    for (int r = 0; r < 8; ++r) acc[ct][r] = bv;
  }

#pragma unroll
  for (int k0 = 0; k0 < CIN; k0 += 4) {
    const int ka = k0 + 2 * half;    // ISA A/B 16x4 lane layout
    v2f a_m = *(const v2f*)(aggRow + ka);
    a_m *= inv;                      // mean aggregation on the fly
    v2f a_x = *(const v2f*)(xRow + ka);
#pragma unroll
    for (int ct = 0; ct < 4; ++ct) {
      const int col = ct * 16 + l16;
      v2f bl, br;
      bl.x = Wl[ka * HID + col];
      bl.y = Wl[(ka + 1) * HID + col];
      br.x = Wr[ka * HID + col];
      br.y = Wr[(ka + 1) * HID + col];
      acc[ct] = __builtin_amdgcn_wmma_f32_16x16x4_f32(
          false, a_m, false, bl, (short)0, acc[ct], false, false);
      acc[ct] = __builtin_amdgcn_wmma_f32_16x16x4_f32(
          false, a_x, false, br, (short)0, acc[ct], false, false);
    }
  }

  // L2 row normalization. C/D layout: VGPR r = row m0+8*half+r, col = lane%16
  // (+16*ct). Reduce squares across the 16 lanes sharing a row (xor 1,2,4,8
  // never crosses the 16-lane halves).
#pragma unroll
  for (int r = 0; r < 8; ++r) {
    float nsq = 0.f;
#pragma unroll
    for (int ct = 0; ct < 4; ++ct) nsq += acc[ct][r] * acc[ct][r];
    nsq += __shfl_xor(nsq, 1);
    nsq += __shfl_xor(nsq, 2);
    nsq += __shfl_xor(nsq, 4);
    nsq += __shfl_xor(nsq, 8);
    const float scl = 1.0f / fmaxf(sqrtf(nsq), 1e-12f);
    const int row = m0 + 8 * half + r;
    if (row < N) {
#pragma unroll
      for (int ct = 0; ct < 4; ++ct) {
        float v = acc[ct][r] * scl;
        if (RELU) v = fmaxf(v, 0.0f);
        out[(long long)row * HID + ct * 16 + l16] = v;
      }
    }
  }
}

// -------------------------------------------------- classifier head (tiny)
__global__ void classifier_kernel(const float* __restrict__ h,
                                  const float* __restrict__ Wc1,
                                  const float* __restrict__ bc1,
                                  const float* __restrict__ Wc2,
                                  const float* __restrict__ bc2,
                                  float* __restrict__ out, int N) {
  int n = blockIdx.x * blockDim.x + threadIdx.x;
  if (n >= N) return;
  float hr[HID];
#pragma unroll
  for (int k = 0; k < HID; ++k) hr[k] = h[(long long)n * HID + k];
  float logit = bc2[0];
#pragma unroll 4
  for (int j = 0; j < 32; ++j) {
    float z = bc1[j];
#pragma unroll
    for (int k = 0; k < HID; ++k) z = fmaf(hr[k], Wc1[k * 32 + j], z);
    logit = fmaf(fmaxf(z, 0.0f), Wc2[j], logit);
  }
  out[n] = logit;
}

// ---------------------------------------------------------------- launch
extern "C" void kernel_launch(void* const* d_in, const int* in_sizes, int n_in,
                              void* d_out, int out_size, void* d_ws, size_t ws_size,
                              hipStream_t stream) {
  const float* x   = (const float*)d_in[0];
  const int*   ei  = (const int*)d_in[1];   // [2, E] flat: row0 = src, row1 = dst
  const float* Wl0 = (const float*)d_in[2];
  const float* b0  = (const float*)d_in[3];
  const float* Wr0 = (const float*)d_in[4];
  const float* Wl1 = (const float*)d_in[5];
  const float* b1  = (const float*)d_in[6];
  const float* Wr1 = (const float*)d_in[7];
  const float* Wc1 = (const float*)d_in[8];
  const float* bc1 = (const float*)d_in[9];
  const float* Wc2 = (const float*)d_in[10];
  const float* bc2 = (const float*)d_in[11];

  const int       N = in_sizes[0] / 32;     // 50000
  const long long E = in_sizes[1] / 2;      // 1600000
  const int* src = ei;
  const int* dst = ei + E;

  // workspace layout (padded to 64-row multiples, all fp32)
  const long long NP = ((N + 63) / 64) * 64LL;
  float* ws   = (float*)d_ws;
  float* cnt  = ws;                 // NP
  float* agg0 = cnt + NP;           // NP*32
  float* agg1 = agg0 + NP * 32;     // NP*64
  float* h0   = agg1 + NP * 64;     // NP*64
  float* h1   = h0 + NP * 64;       // NP*64
  (void)ws_size; (void)n_in; (void)out_size;

  // zero cnt + agg0 + agg1 (contiguous prefix) every call
  {
    long long z4 = (NP * (1 + 32 + 64)) / 4;
    zero_kernel<<<(int)((z4 + 255) / 256), 256, 0, stream>>>((float4*)d_ws, z4);
  }

  // layer 0: scatter-mean numerator + counts, then WMMA transform (+ReLU)
  {
    long long total = E * 32;
    scatter_kernel<32, true><<<(int)((total + 255) / 256), 256, 0, stream>>>(
        x, src, dst, agg0, cnt, E);
  }
  sage_wmma_kernel<32, true><<<(N + 127) / 128, 256, 0, stream>>>(
      agg0, cnt, x, Wl0, Wr0, b0, h0, N);

  // layer 1: counts are identical, only re-scatter features
  {
    long long total = E * 64;
    scatter_kernel<64, false><<<(int)((total + 255) / 256), 256, 0, stream>>>(
        h0, src, dst, agg1, nullptr, E);
  }
  sage_wmma_kernel<64, false><<<(N + 127) / 128, 256, 0, stream>>>(
      agg1, cnt, h0, Wl1, Wr1, b1, h1, N);

  // classifier head -> logits
  classifier_kernel<<<(N + 255) / 256, 256, 0, stream>>>(
      h1, Wc1, bc1, Wc2, bc2, (float*)d_out, N);
}